// SimpleModel_59545426592235
// MI455X (gfx1250) — compile-verified
//
#include <hip/hip_runtime.h>
#include <hip/hip_bf16.h>
#include <math.h>

typedef __attribute__((ext_vector_type(2))) float v2f;
typedef __attribute__((ext_vector_type(8))) float v8f;

// ---------------------------------------------------------------------------
// K1: zero workspace region (cnt[N] int, t[2N] f32, acc2[2N] f32 -> 5N words)
// ---------------------------------------------------------------------------
__global__ void zero_kernel(int* __restrict__ p, int n) {
    int i = blockIdx.x * blockDim.x + threadIdx.x;
    if (i < n) p[i] = 0;
}

// ---------------------------------------------------------------------------
// K2: degree count over col (deg = cnt + 1 for self-loop)
// ---------------------------------------------------------------------------
__global__ void deg_kernel(const int* __restrict__ col, int* __restrict__ cnt, int E) {
    int e = blockIdx.x * blockDim.x + threadIdx.x;
    if (e < E) atomicAdd(&cnt[col[e]], 1);
}

// ---------------------------------------------------------------------------
// K3: dinv = rsqrt(cnt+1); xd = x * dinv  (2 floats/node)
// ---------------------------------------------------------------------------
__global__ void dinv_kernel(const int* __restrict__ cnt, const float* __restrict__ x,
                            float* __restrict__ dinv, float* __restrict__ xd, int N) {
    int i = blockIdx.x * blockDim.x + threadIdx.x;
    if (i >= N) return;
    float d = rsqrtf((float)cnt[i] + 1.0f);
    dinv[i] = d;
    xd[2 * i]     = x[2 * i]     * d;
    xd[2 * i + 1] = x[2 * i + 1] * d;
}

// ---------------------------------------------------------------------------
// K4: layer-1 scatter: t[col] += xd[row]
// ---------------------------------------------------------------------------
__global__ void scatter1_kernel(const int* __restrict__ row, const int* __restrict__ col,
                                const float* __restrict__ xd, float* __restrict__ t, int E) {
    int e = blockIdx.x * blockDim.x + threadIdx.x;
    if (e >= E) return;
    int r = row[e], c = col[e];
    atomicAdd(&t[2 * c],     xd[2 * r]);
    atomicAdd(&t[2 * c + 1], xd[2 * r + 1]);
}

// ---------------------------------------------------------------------------
// Pack kernel (1 block / 256 threads): per-lane-contiguous operand tables.
//   i = lane*8 + j,  j = kk*2 + which,  K/col = 4*kk + 2*hi + which
//   pk[   0+i] = packed B  [We_top | We_bot | W2 | 0]  value B[K][n]
//   pk[ 256+i] = W1 row0 [col]
//   pk[ 512+i] = W1 row1 [col]
//   pk[ 768+i] = b1      [col]
// ---------------------------------------------------------------------------
__global__ void pack_kernel(const float* __restrict__ W1, const float* __restrict__ b1,
                            const float* __restrict__ We, const float* __restrict__ W2,
                            float* __restrict__ pk) {
    int i    = threadIdx.x;            // 0..255
    int lane = i >> 3;
    int j    = i & 7;
    int hi   = lane >> 4;
    int n    = lane & 15;
    int col  = 4 * (j >> 1) + 2 * hi + (j & 1);
    float bv;
    if (n < 4)       bv = We[col * 4 + n];
    else if (n < 8)  bv = We[(17 + col) * 4 + (n - 4)];
    else if (n < 10) bv = W2[col * 2 + (n - 8)];
    else             bv = 0.0f;
    pk[i]       = bv;
    pk[256 + i] = W1[col];
    pk[512 + i] = W1[16 + col];
    pk[768 + i] = b1[col];
}

// ---------------------------------------------------------------------------
// K5: WMMA node kernel. One wave32 handles 16 nodes.
//   h[m] = dinv[m]*((t[m]+xd[m]) @ W1) + b1      (built directly in A layout)
//   D = h @ [We_top | We_bot | W2 | 0]  via 4 chained v_wmma_f32_16x16x4_f32
//   stores (branch-free): a = D[:,0:4]+be, c = D[:,4:8], s2 = D[:,8:10]*dinv
// ---------------------------------------------------------------------------
__global__ void node_wmma_kernel(const float* __restrict__ t, const float* __restrict__ xd,
                                 const float* __restrict__ dinv,
                                 const float* __restrict__ pk, const float* __restrict__ be,
                                 float* __restrict__ wsf, int N,
                                 int offA, int offC, int offS, int dumpOff) {
    int wave = blockIdx.x * (blockDim.x >> 5) + (threadIdx.x >> 5);
    int base = wave * 16;
    if (base >= N) return;            // wave-uniform: EXEC stays all-1s for WMMA

    int lane = threadIdx.x & 31;
    int hi   = lane >> 4;
    int n    = lane & 15;

    int m = base + n; if (m >= N) m = N - 1;   // clamp (padded rows never stored)
    float  d  = dinv[m];
    float2 tm = *(const float2*)(t  + 2 * m);
    float2 xm = *(const float2*)(xd + 2 * m);
    float  u0 = tm.x + xm.x;
    float  u1 = tm.y + xm.y;

    // per-lane operand tables: 8 straight b128 loads, no divergence
    const float4* p4 = (const float4*)pk + lane * 2;
    float4 bB0 = p4[0],   bB1 = p4[1];     // packed B, chunks 0..3
    float4 w00 = p4[64],  w01 = p4[65];    // W1 row0
    float4 w10 = p4[128], w11 = p4[129];   // W1 row1
    float4 bb0 = p4[192], bb1 = p4[193];   // b1

    v8f acc = {0.f, 0.f, 0.f, 0.f, 0.f, 0.f, 0.f, 0.f};
    v2f A, B;
    // chunk 0
    A.x = d * (u0 * w00.x + u1 * w10.x) + bb0.x;
    A.y = d * (u0 * w00.y + u1 * w10.y) + bb0.y;
    B.x = bB0.x; B.y = bB0.y;
    acc = __builtin_amdgcn_wmma_f32_16x16x4_f32(false, A, false, B, (short)0, acc, false, false);
    // chunk 1
    A.x = d * (u0 * w00.z + u1 * w10.z) + bb0.z;
    A.y = d * (u0 * w00.w + u1 * w10.w) + bb0.w;
    B.x = bB0.z; B.y = bB0.w;
    acc = __builtin_amdgcn_wmma_f32_16x16x4_f32(false, A, false, B, (short)0, acc, false, false);
    // chunk 2
    A.x = d * (u0 * w01.x + u1 * w11.x) + bb1.x;
    A.y = d * (u0 * w01.y + u1 * w11.y) + bb1.y;
    B.x = bB1.x; B.y = bB1.y;
    acc = __builtin_amdgcn_wmma_f32_16x16x4_f32(false, A, false, B, (short)0, acc, false, false);
    // chunk 3
    A.x = d * (u0 * w01.z + u1 * w11.z) + bb1.z;
    A.y = d * (u0 * w01.w + u1 * w11.w) + bb1.w;
    B.x = bB1.z; B.y = bB1.w;
    acc = __builtin_amdgcn_wmma_f32_16x16x4_f32(false, A, false, B, (short)0, acc, false, false);

    // ---- fully branch-free store phase -------------------------------
    // D layout: acc[j] = D[M = j + 8*hi][N = n]
    float beval = be[n & 3];                          // always in-range
    bool isA   = (n < 4);
    bool isS2  = (n >= 8) && (n < 10);
    bool lives = (n < 10);
    float add  = isA ? beval : 0.0f;
    int off0   = isA ? (offA + n)
               : (n < 8) ? (offC + (n - 4))
               :           (offS + (n & 1));
    int stride = isS2 ? 2 : 4;

#pragma unroll
    for (int j = 0; j < 8; ++j) {
        int   mm = base + j + 8 * hi;
        float dv = __shfl(d, 8 * hi + j, 32);         // dinv[mm] via bpermute
        float v  = acc[j];
        v = isS2 ? v * dv : v + add;
        bool ok  = lives && (mm < N);
        int  idx = ok ? (off0 + mm * stride) : (dumpOff + lane);
        wsf[idx] = v;                                 // unconditional store
    }
}

// ---------------------------------------------------------------------------
// K6: fused edge pass: edge MLP (relu + log_softmax over 4) + layer-2 scatter
// ---------------------------------------------------------------------------
__global__ void edge_mlp_kernel(const int* __restrict__ row, const int* __restrict__ col,
                                const float* __restrict__ ea, const float* __restrict__ We,
                                const float* __restrict__ aout, const float* __restrict__ cout,
                                const float* __restrict__ s2, float* __restrict__ acc2,
                                float* __restrict__ eout, int E) {
    int e = blockIdx.x * blockDim.x + threadIdx.x;
    if (e >= E) return;
    int r = row[e], c = col[e];
    float4 av = *(const float4*)(aout + 4 * r);
    float4 cv = *(const float4*)(cout + 4 * c);
    float xa = ea[e];
    float v0 = av.x + cv.x + xa * We[64];
    float v1 = av.y + cv.y + xa * We[65];
    float v2 = av.z + cv.z + xa * We[66];
    float v3 = av.w + cv.w + xa * We[67];
    v0 = fmaxf(v0, 0.f); v1 = fmaxf(v1, 0.f);
    v2 = fmaxf(v2, 0.f); v3 = fmaxf(v3, 0.f);
    float mx  = fmaxf(fmaxf(v0, v1), fmaxf(v2, v3));
    float s   = expf(v0 - mx) + expf(v1 - mx) + expf(v2 - mx) + expf(v3 - mx);
    float lse = mx + logf(s);
    float4 o  = make_float4(v0 - lse, v1 - lse, v2 - lse, v3 - lse);
    *(float4*)(eout + 4 * e) = o;
    atomicAdd(&acc2[2 * c],     s2[2 * r]);
    atomicAdd(&acc2[2 * c + 1], s2[2 * r + 1]);
}

// ---------------------------------------------------------------------------
// K7: out_x = dinv*(acc2 + s2) + b2, then log_softmax over 2
// ---------------------------------------------------------------------------
__global__ void final_node_kernel(const float* __restrict__ acc2, const float* __restrict__ s2,
                                  const float* __restrict__ dinv, const float* __restrict__ b2,
                                  float* __restrict__ out, int N) {
    int i = blockIdx.x * blockDim.x + threadIdx.x;
    if (i >= N) return;
    float d  = dinv[i];
    float o0 = d * (acc2[2 * i]     + s2[2 * i])     + b2[0];
    float o1 = d * (acc2[2 * i + 1] + s2[2 * i + 1]) + b2[1];
    float mx  = fmaxf(o0, o1);
    float lse = mx + logf(expf(o0 - mx) + expf(o1 - mx));
    out[2 * i]     = o0 - lse;
    out[2 * i + 1] = o1 - lse;
}

// ---------------------------------------------------------------------------
extern "C" void kernel_launch(void* const* d_in, const int* in_sizes, int n_in,
                              void* d_out, int out_size, void* d_ws, size_t ws_size,
                              hipStream_t stream) {
    const float* x  = (const float*)d_in[0];
    const int*   ei = (const int*)  d_in[1];
    const float* ea = (const float*)d_in[2];
    const float* W1 = (const float*)d_in[3];
    const float* b1 = (const float*)d_in[4];
    const float* We = (const float*)d_in[5];
    const float* be = (const float*)d_in[6];
    const float* W2 = (const float*)d_in[7];
    const float* b2 = (const float*)d_in[8];

    const int N = in_sizes[0] / 2;
    const int E = in_sizes[1] / 2;
    const int* row = ei;
    const int* col = ei + E;

    // ws layout (floats): cnt[N] t[2N] acc2[2N] dinv[N] xd[2N] a[4N] c[4N] s2[2N] pk[1024] dump[32]
    float* ws   = (float*)d_ws;
    int*   cnt  = (int*)ws;                  // [0,  N)
    float* t    = ws + (size_t)N;            // [N, 3N)
    float* acc2 = ws + (size_t)3 * N;        // [3N,5N)
    float* dinv = ws + (size_t)5 * N;
    float* xd   = ws + (size_t)6 * N;
    const int offA = 8 * N;                  // 32N bytes -> 16B aligned
    const int offC = 12 * N;
    const int offS = 16 * N;
    float* aout = ws + (size_t)offA;
    float* cout = ws + (size_t)offC;
    float* s2   = ws + (size_t)offS;
    size_t pko  = (((size_t)18 * N) + 3) & ~(size_t)3;   // 16B align
    float* pk   = ws + pko;
    const int dumpOff = (int)pko + 1024;     // 32-float dump area after pk
    float* out  = (float*)d_out;             // [0,2N) nodes, [2N,2N+4E) edges

    const int B = 256;
    int zb   = (5 * N + B - 1) / B;
    int nb   = (N + B - 1) / B;
    int ebk  = (E + B - 1) / B;
    int waves = (N + 15) / 16;
    int wblks = (waves + 7) / 8;             // 8 waves (256 threads) per block

    zero_kernel<<<zb, B, 0, stream>>>(cnt, 5 * N);
    pack_kernel<<<1, 256, 0, stream>>>(W1, b1, We, W2, pk);
    deg_kernel<<<ebk, B, 0, stream>>>(col, cnt, E);
    dinv_kernel<<<nb, B, 0, stream>>>(cnt, x, dinv, xd, N);
    scatter1_kernel<<<ebk, B, 0, stream>>>(row, col, xd, t, E);
    node_wmma_kernel<<<wblks, B, 0, stream>>>(t, xd, dinv, pk, be, ws, N,
                                              offA, offC, offS, dumpOff);
    edge_mlp_kernel<<<ebk, B, 0, stream>>>(row, col, ea, We, aout, cout, s2,
                                           acc2, out + (size_t)2 * N, E);
    final_node_kernel<<<nb, B, 0, stream>>>(acc2, s2, dinv, b2, out, N);
}